// LSTMModel_35390530519091
// MI455X (gfx1250) — compile-verified
//
#include <hip/hip_runtime.h>

// MI455X fused LSTM + Linear (fp32 end-to-end)
//   phase 1: per-thread LSTM (128 rows/WG), hs kept in LDS; idle waves
//            concurrently stage W-tile 0 and b_lin
//   phase 2: fp32 WMMA (16x16x4) GEMM out[128x672] = hs[128x288] @ W_lin^T,
//            bias folded into the WMMA C operand, W tiles double-buffered and
//            staged with GLOBAL_LOAD_ASYNC_TO_LDS_B64 (ASYNCcnt), LDS-only
//            split barriers (no per-tile storecnt drain).
// HBM traffic = x (88 MB) + out (88 MB); hs never leaves the WGP, W_lin is
// L2-resident (0.77 MB).

typedef float v2f __attribute__((ext_vector_type(2)));
typedef float v8f __attribute__((ext_vector_type(8)));
typedef int   v2i __attribute__((__vector_size__(8)));

#define Tt        96
#define KH        288          // GEMM K = T*H
#define NG        672          // GEMM N = T*O (also x/out row length)
#define MB        128          // batch rows per workgroup
#define HS_PITCH  290          // even (8B-aligned b64 frags) + conflict-free
#define NT        42           // NG/16
#define KK        72           // KH/4
#define WT_FLOATS (144*32)     // 288x16 W tile, pair-interleaved for b64 frags
#define THREADS   256          // 8 waves: 1 M-subtile each

#define AS1 __attribute__((address_space(1)))
#define AS3 __attribute__((address_space(3)))

#if __has_builtin(__builtin_amdgcn_global_load_async_to_lds_b64)
#define HAVE_ASYNC_LDS 1
#else
#define HAVE_ASYNC_LDS 0
#endif

__device__ __forceinline__ float fsigmoid(float x) {
    float e = __expf(-x);                       // v_exp_f32
    return __builtin_amdgcn_rcpf(1.0f + e);     // v_rcp_f32
}
__device__ __forceinline__ float ftanh_fast(float x) {
    float e = __expf(-2.0f * x);                // tanh = 2/(1+e^-2x) - 1
    return __builtin_amdgcn_rcpf(1.0f + e) * 2.0f - 1.0f;
}
__device__ __forceinline__ v2f mk2(float a, float b) {
    v2f r; r.x = a; r.y = b; return r;
}

// LDS-only split barrier: drains DS + async-LDS traffic but NOT the global
// output stores (unlike __syncthreads, which emits s_wait_storecnt 0).
__device__ __forceinline__ void wg_barrier_lds() {
    asm volatile(
        "s_wait_dscnt 0x0\n\t"
        "s_wait_asynccnt 0x0\n\t"
        "s_barrier_signal -1\n\t"
        "s_barrier_wait -1" ::: "memory");
}

// Stage one 16x288 W_lin tile into LDS, pre-swizzled into B-fragment pair
// order: element (jj,k) -> dst[(k>>1)*32 + jj*2 + (k&1)].
__device__ __forceinline__ void stage_wtile(const float* __restrict__ Wb,
                                            float* __restrict__ dst,
                                            int tid, int stride) {
#if HAVE_ASYNC_LDS
    // even/odd k land in consecutive dwords -> one async b64 per dword-pair
    for (int p = tid; p < (16 * KH) / 2; p += stride) {
        int jj = p / 144;           // N row within tile
        int m  = p - jj * 144;      // k-pair index
        __builtin_amdgcn_global_load_async_to_lds_b64(
            (AS1 v2i*)(Wb + jj * KH + 2 * m),
            (AS3 v2i*)(dst + m * 32 + 2 * jj), 0, 0);
    }
#else
    for (int i = tid; i < 16 * KH; i += stride) {
        int jj = i / KH; int k = i - jj * KH;
        dst[(k >> 1) * 32 + jj * 2 + (k & 1)] = Wb[i];
    }
#endif
}

__global__ void __launch_bounds__(THREADS, 1)
lstm_linear_fused(const float* __restrict__ x,
                  const float* __restrict__ W_ih,
                  const float* __restrict__ W_hh,
                  const float* __restrict__ b_ih,
                  const float* __restrict__ b_hh,
                  const float* __restrict__ W_lin,
                  const float* __restrict__ b_lin,
                  float* __restrict__ out)
{
    extern __shared__ float smem[];
    float* hsb  = smem;                        // [128][290]
    float* wt0  = smem + MB * HS_PITCH;        // W tile ping
    float* wt1  = wt0 + WT_FLOATS;             // W tile pong
    float* blin = wt1 + WT_FLOATS;             // [672]

    const int tid    = threadIdx.x;
    const int wgBase = blockIdx.x * MB;

    // ---------------- Phase 1: LSTM over T=96 (waves 0-3) ----------------
    if (tid < MB) {
        // gate pairs packed for V_PK_FMA_F32 (2 FMA/instr)
        v2f Wih2[6][7], Whh2[6][3], bias2[6];
        #pragma unroll
        for (int p = 0; p < 6; ++p) {
            #pragma unroll
            for (int d = 0; d < 7; ++d)
                Wih2[p][d] = mk2(W_ih[(2*p)*7 + d], W_ih[(2*p+1)*7 + d]);
            #pragma unroll
            for (int q = 0; q < 3; ++q)
                Whh2[p][q] = mk2(W_hh[(2*p)*3 + q], W_hh[(2*p+1)*3 + q]);
            bias2[p] = mk2(b_ih[2*p]   + b_hh[2*p],
                           b_ih[2*p+1] + b_hh[2*p+1]);
        }
        const float* xb = x + (size_t)(wgBase + tid) * NG;
        float* hrow = hsb + tid * HS_PITCH;    // conflict-free (290 % 64 = 34)
        float h[3] = {0.f,0.f,0.f}, c[3] = {0.f,0.f,0.f};
        for (int t = 0; t < Tt; ++t) {
            float xv[7];
            #pragma unroll
            for (int d = 0; d < 7; ++d) xv[d] = xb[t*7 + d];
            v2f acc2[6];
            #pragma unroll
            for (int p = 0; p < 6; ++p) acc2[p] = bias2[p];
            #pragma unroll
            for (int d = 0; d < 7; ++d) {
                v2f xs = mk2(xv[d], xv[d]);
                #pragma unroll
                for (int p = 0; p < 6; ++p)
                    acc2[p] = __builtin_elementwise_fma(xs, Wih2[p][d], acc2[p]);
            }
            #pragma unroll
            for (int q = 0; q < 3; ++q) {
                v2f hq = mk2(h[q], h[q]);
                #pragma unroll
                for (int p = 0; p < 6; ++p)
                    acc2[p] = __builtin_elementwise_fma(hq, Whh2[p][q], acc2[p]);
            }
            float ac[12];
            #pragma unroll
            for (int p = 0; p < 6; ++p) { ac[2*p] = acc2[p].x; ac[2*p+1] = acc2[p].y; }
            #pragma unroll
            for (int q = 0; q < 3; ++q) {       // gate order i,f,g,o
                float ig = fsigmoid(ac[q]);
                float fg = fsigmoid(ac[3+q]);
                float gg = ftanh_fast(ac[6+q]);
                float og = fsigmoid(ac[9+q]);
                c[q] = __fmaf_rn(fg, c[q], ig * gg);
                h[q] = og * ftanh_fast(c[q]);
                hrow[t*3 + q] = h[q];
            }
        }
    } else {
        // waves 4-7: stage W tile 0 + b_lin while the LSTM runs
        const int t2 = tid - MB;
        stage_wtile(W_lin, wt0, t2, THREADS - MB);
        for (int i = t2; i < NG; i += THREADS - MB) blin[i] = b_lin[i];
    }
    // drain any async staging before the full phase barrier
    asm volatile("s_wait_asynccnt 0x0" ::: "memory");
    __syncthreads();

    // ------------- Phase 2: WMMA GEMM, fp32 16x16x4, 8 waves -------------
    const int lane = tid & 31;
    const int wid  = tid >> 5;      // 8 waves -> one 16-row M subtile each
    const int j    = lane & 15;
    const int hi   = lane >> 4;
    const int m0   = wid * 16;
    // A frag (16x4 f32): VGPR0 = K{0|2}, VGPR1 = K{1|3} -> one b64 per frag
    const v2f* Ap   = (const v2f*)(hsb + (m0 + j) * HS_PITCH + 2*hi);
    const int  bOff = hi*16 + j;    // B frag base in pair-interleaved tile

    float bl = blin[j];             // bias for tile 0
    for (int n0 = 0; n0 < NT; ++n0) {
        float* cur = (n0 & 1) ? wt1 : wt0;
        float* nxt = (n0 & 1) ? wt0 : wt1;
        if (n0 + 1 < NT)            // async-stage tile n0+1 during compute
            stage_wtile(W_lin + (size_t)(n0 + 1) * 16 * KH, nxt, tid, THREADS);

        // prefetch next tile's bias; current one folds into the accumulator
        float blNext = blin[((n0 + 1 < NT) ? (n0 + 1) : n0) * 16 + j];
        v8f  cc = {bl, bl, bl, bl, bl, bl, bl, bl};   // D = A*B + bias
        const v2f* Bp = (const v2f*)cur + bOff;
        #pragma unroll 8
        for (int kk = 0; kk < KK; ++kk) {
            v2f a  = Ap[2*kk];           // ds_load_b64, conflict-free
            v2f bb = Bp[32*kk];
            cc = __builtin_amdgcn_wmma_f32_16x16x4_f32(
                     false, a, false, bb, (short)0, cc, false, false);
        }

        // C/D layout: VGPR v, lanes0-15 -> (M=v, N=j); lanes16-31 -> (M=v+8, N=j)
        float* o = out + (size_t)(wgBase + m0 + hi*8) * NG + n0*16 + j;
        #pragma unroll
        for (int v = 0; v < 8; ++v)
            o[(size_t)v * NG] = cc[v];

        bl = blNext;
        wg_barrier_lds();           // LDS visibility only; stores drain async
    }
}

extern "C" void kernel_launch(void* const* d_in, const int* in_sizes, int n_in,
                              void* d_out, int out_size, void* d_ws, size_t ws_size,
                              hipStream_t stream) {
    (void)in_sizes; (void)n_in; (void)d_ws; (void)ws_size; (void)out_size;
    const float* x     = (const float*)d_in[0];
    const float* W_ih  = (const float*)d_in[1];
    const float* W_hh  = (const float*)d_in[2];
    const float* b_ih  = (const float*)d_in[3];
    const float* b_hh  = (const float*)d_in[4];
    const float* W_lin = (const float*)d_in[5];
    const float* b_lin = (const float*)d_in[6];
    float* out = (float*)d_out;

    const int    blocks = 32768 / MB;  // 256
    const size_t shmem  = (MB * HS_PITCH + 2 * WT_FLOATS + NG) * sizeof(float); // 188,032 B (<320KB/WG)
    lstm_linear_fused<<<blocks, THREADS, shmem, stream>>>(
        x, W_ih, W_hh, b_ih, b_hh, W_lin, b_lin, out);
}